// RelationModule_59322088293188
// MI455X (gfx1250) — compile-verified
//
#include <hip/hip_runtime.h>
#include <math.h>

// ---------------------------------------------------------------------------
// RelationModule on MI455X (gfx1250, wave32).
// Memory-bound on the 256MB position_embedding read; all GEMMs done with
// V_WMMA_F32_16X16X4_F32 so compute never limits.  out = att @ (nongt@Wc^T)
// refactoring cuts 34 GFLOP -> 4.2 GFLOP for the attention-apply stage.
// L2 policy (192MB global L2): the 256MB read-once embedding stream is loaded
// non-temporal so the 64MB plog intermediate stays L2-resident between
// k_poslog (producer) and k_attn (consumer); read-once / write-once traffic
// (plog in k_attn, final out) is likewise NT.
// ---------------------------------------------------------------------------

#define FEAT 1024
#define EMB  64
#define GROUP 16
#define DG   64
#define RROI 1024
#define NN   1024

typedef float v2f __attribute__((ext_vector_type(2)));
typedef float v8f __attribute__((ext_vector_type(8)));

// D = A(16x4) x B(4x16) + C(16x16), f32, wave32.
// Lane layout: lanes 0-15 hold K=0,1 (A: M=lane, B: N=lane); lanes 16-31 hold K=2,3.
// C/D: VGPR j = row j (lanes 0-15) / row j+8 (lanes 16-31), col = lane&15.
__device__ __forceinline__ v8f wmma_f32(v2f a, v2f b, v8f c) {
  return __builtin_amdgcn_wmma_f32_16x16x4_f32(
      /*neg_a=*/false, a, /*neg_b=*/false, b,
      /*c_mod=*/(short)0, c, /*reuse_a=*/false, /*reuse_b=*/false);
}

__device__ __forceinline__ float half_reduce_max(float v) {
  v = fmaxf(v, __shfl_xor(v, 1, 16));
  v = fmaxf(v, __shfl_xor(v, 2, 16));
  v = fmaxf(v, __shfl_xor(v, 4, 16));
  v = fmaxf(v, __shfl_xor(v, 8, 16));
  return v;
}
__device__ __forceinline__ float half_reduce_sum(float v) {
  v += __shfl_xor(v, 1, 16);
  v += __shfl_xor(v, 2, 16);
  v += __shfl_xor(v, 4, 16);
  v += __shfl_xor(v, 8, 16);
  return v;
}

// ---------------------------------------------------------------------------
// Kernel 1: q = roi@Wq^T + bq and k = roi@Wk^T + bk  (both 1024x1024, K=1024)
// Wave computes a 16(M) x 64(N) tile; A fragment reused across 4 N-subtiles.
// roi/Wq/Wk are 4MB each and heavily reused -> default (RT) caching.
// ---------------------------------------------------------------------------
__global__ __launch_bounds__(256) void k_proj(
    const float* __restrict__ X,
    const float* __restrict__ Wq, const float* __restrict__ bq,
    const float* __restrict__ Wk, const float* __restrict__ bk,
    float* __restrict__ qout, float* __restrict__ kout)
{
  const int lane = threadIdx.x & 31;
  const int wave = threadIdx.x >> 5;
  const int wid  = blockIdx.x * 8 + wave;   // 0..2047
  const int mat  = wid >> 10;               // 0 -> q, 1 -> k
  const int rem  = wid & 1023;
  const int m0   = (rem >> 4) * 16;
  const int n0   = (rem & 15) * 64;
  const float* W = mat ? Wk : Wq;
  const float* b = mat ? bk : bq;
  float*       Y = mat ? kout : qout;

  const int hl = lane >> 4;
  const int ll = lane & 15;
  const float* xrow = X + (m0 + ll) * FEAT + 2 * hl;
  const float* wrow = W + (n0 + ll) * FEAT + 2 * hl;

  v8f c0 = {}, c1 = {}, c2 = {}, c3 = {};
#pragma unroll 4
  for (int k0 = 0; k0 < FEAT; k0 += 4) {
    v2f a  = *(const v2f*)(xrow + k0);
    c0 = wmma_f32(a, *(const v2f*)(wrow + k0),             c0);
    c1 = wmma_f32(a, *(const v2f*)(wrow + 16 * FEAT + k0), c1);
    c2 = wmma_f32(a, *(const v2f*)(wrow + 32 * FEAT + k0), c2);
    c3 = wmma_f32(a, *(const v2f*)(wrow + 48 * FEAT + k0), c3);
  }
  v8f cs[4] = {c0, c1, c2, c3};
#pragma unroll
  for (int t = 0; t < 4; ++t) {
    float bb = b[n0 + t * 16 + ll];
    v8f c = cs[t];
#pragma unroll
    for (int j = 0; j < 8; ++j) {
      int row = m0 + j + 8 * hl;
      Y[row * FEAT + n0 + t * 16 + ll] = c[j] + bb;
    }
  }
}

// ---------------------------------------------------------------------------
// Kernel 2: KVt[g][o][n] = sum_f nongt[n][f] * Wc[g][o][f]   (transposed store
// so the attention kernel's B fragments are contiguous float2 along n).
// ---------------------------------------------------------------------------
__global__ __launch_bounds__(256) void k_kv(
    const float* __restrict__ nongt, const float* __restrict__ Wc,
    float* __restrict__ KVt)
{
  const int lane = threadIdx.x & 31;
  const int wave = threadIdx.x >> 5;
  const int wid  = blockIdx.x * 8 + wave;  // 0..1023
  const int g    = wid >> 6;
  const int n0   = (wid & 63) * 16;
  const int hl = lane >> 4, ll = lane & 15;

  const float* arow = nongt + (n0 + ll) * FEAT + 2 * hl;
  const float* wrow = Wc + (size_t)(g * DG + ll) * FEAT + 2 * hl;

  v8f c0 = {}, c1 = {}, c2 = {}, c3 = {};
#pragma unroll 2
  for (int k0 = 0; k0 < FEAT; k0 += 4) {
    v2f a = *(const v2f*)(arow + k0);
    c0 = wmma_f32(a, *(const v2f*)(wrow + k0),             c0);
    c1 = wmma_f32(a, *(const v2f*)(wrow + 16 * FEAT + k0), c1);
    c2 = wmma_f32(a, *(const v2f*)(wrow + 32 * FEAT + k0), c2);
    c3 = wmma_f32(a, *(const v2f*)(wrow + 48 * FEAT + k0), c3);
  }
  v8f cs[4] = {c0, c1, c2, c3};
#pragma unroll
  for (int t = 0; t < 4; ++t) {
    v8f c = cs[t];
    float* dst = KVt + (size_t)(g * DG + t * 16 + ll) * NN + n0 + 8 * hl;
    *(float4*)(dst)     = make_float4(c[0], c[1], c[2], c[3]);
    *(float4*)(dst + 4) = make_float4(c[4], c[5], c[6], c[7]);
  }
}

// ---------------------------------------------------------------------------
// Kernel 3 (bandwidth king, 256MB read-once): plog[r][g][n] =
//   log(max(relu(pos_emb[r][n]·Wp[g] + bp[g]), 1e-6)).  Wp B-fragments live
// entirely in registers; embedding loads are NON-TEMPORAL (stream, don't
// evict L2); output written in the (r,g,n) layout k_attn streams, via
// coalesced float4 stores with default (RT) policy so it parks in L2.
// ---------------------------------------------------------------------------
__global__ __launch_bounds__(256) void k_poslog(
    const float* __restrict__ pe, const float* __restrict__ Wp,
    const float* __restrict__ bp, float* __restrict__ plog)
{
  const int lane = threadIdx.x & 31;
  const int wave = threadIdx.x >> 5;
  const int tile = blockIdx.x * 8 + wave;  // 0..65535
  const int r  = tile >> 6;
  const int n0 = (tile & 63) * 16;
  const int hl = lane >> 4, ll = lane & 15;

  v2f bf[16];
  const float* wprow = Wp + ll * EMB + 2 * hl;
#pragma unroll
  for (int s = 0; s < 16; ++s) bf[s] = *(const v2f*)(wprow + 4 * s);

  const float* arow = pe + (size_t)(r * NN + n0 + ll) * EMB + 2 * hl;
  __builtin_prefetch(arow + 16 * EMB, 0, 0);  // next n-tile of this row

  v8f c = {};
#pragma unroll
  for (int s = 0; s < 16; ++s) {
    v2f a = __builtin_nontemporal_load((const v2f*)(arow + 4 * s));
    c = wmma_f32(a, bf[s], c);
  }
  float bpv = bp[ll];
  float o[8];
#pragma unroll
  for (int j = 0; j < 8; ++j) o[j] = __logf(fmaxf(c[j] + bpv, 1e-6f));

  float* dst = plog + (size_t)(r * GROUP + ll) * NN + n0 + 8 * hl;
  *(float4*)(dst)     = make_float4(o[0], o[1], o[2], o[3]);
  *(float4*)(dst + 4) = make_float4(o[4], o[5], o[6], o[7]);
}

// ---------------------------------------------------------------------------
// Kernel 4: fused aff + log-weight + online softmax + att@KV + bias.
// One wave per (16-row r-tile, group g).  p-tile is re-fragmented through
// padded LDS (stride 18 -> bank-conflict-free) to become the A matrix of the
// second WMMA GEMM.  Row statistics use 16-lane half-wave shfl reductions,
// matching the WMMA C-layout (row j+8h lives in half h).  plog is read-once
// -> non-temporal; q/k/KVt are L2-resident and reused -> default policy.
// ---------------------------------------------------------------------------
__global__ __launch_bounds__(256) void k_attn(
    const float* __restrict__ q, const float* __restrict__ kmat,
    const float* __restrict__ KVt, const float* __restrict__ plog,
    const float* __restrict__ bc, float* __restrict__ out)
{
  __shared__ float plds[8][16 * 18];
  const int lane = threadIdx.x & 31;
  const int wave = threadIdx.x >> 5;
  const int wid  = blockIdx.x * 8 + wave;  // 0..1023
  const int g    = wid & 15;
  const int r0   = (wid >> 4) * 16;
  const int hl = lane >> 4, ll = lane & 15;
  const float scale = 0.125f;  // 1/sqrt(64)

  v2f qa[16];
  const float* qrow = q + (r0 + ll) * FEAT + g * DG + 2 * hl;
#pragma unroll
  for (int s = 0; s < 16; ++s) qa[s] = *(const v2f*)(qrow + 4 * s);

  v8f acc0 = {}, acc1 = {}, acc2 = {}, acc3 = {};
  float mrow[8], lrow[8];
#pragma unroll
  for (int j = 0; j < 8; ++j) { mrow[j] = -3.0e38f; lrow[j] = 0.f; }

  float* myl = plds[wave];

  for (int n0 = 0; n0 < NN; n0 += 16) {
    // ---- aff tile: q (16xDG) x k^T (DGx16), K = 64 -> 16 WMMAs ----
    v8f c = {};
    const float* krow = kmat + (n0 + ll) * FEAT + g * DG + 2 * hl;
#pragma unroll
    for (int s = 0; s < 16; ++s)
      c = wmma_f32(qa[s], *(const v2f*)(krow + 4 * s), c);

    // ---- log-weight add + online softmax ----
#pragma unroll
    for (int j = 0; j < 8; ++j) {
      int r  = r0 + j + 8 * hl;
      float w = __builtin_nontemporal_load(
                    plog + (size_t)(r * GROUP + g) * NN + n0 + ll);
      float v = c[j] * scale + w;
      float mn   = fmaxf(mrow[j], half_reduce_max(v));
      float corr = __expf(mrow[j] - mn);
      float p    = __expf(v - mn);
      lrow[j] = lrow[j] * corr + half_reduce_sum(p);
      mrow[j] = mn;
      acc0[j] *= corr; acc1[j] *= corr; acc2[j] *= corr; acc3[j] *= corr;
      myl[(j + 8 * hl) * 18 + ll] = p;   // p-tile (row-major, padded)
    }

    // ---- second GEMM: p (16x16) x KVt-tile (16x64), K = 16 -> 16 WMMAs ----
    v2f pa[4];
#pragma unroll
    for (int s = 0; s < 4; ++s)
      pa[s] = *(const v2f*)(&myl[ll * 18 + 4 * s + 2 * hl]);
    const float* kvrow = KVt + (size_t)(g * DG + ll) * NN + n0 + 2 * hl;
#pragma unroll
    for (int s = 0; s < 4; ++s) {
      acc0 = wmma_f32(pa[s], *(const v2f*)(kvrow + 4 * s),             acc0);
      acc1 = wmma_f32(pa[s], *(const v2f*)(kvrow + 16 * NN + 4 * s),   acc1);
      acc2 = wmma_f32(pa[s], *(const v2f*)(kvrow + 32 * NN + 4 * s),   acc2);
      acc3 = wmma_f32(pa[s], *(const v2f*)(kvrow + 48 * NN + 4 * s),   acc3);
    }
  }

  // ---- epilogue: divide by softmax denom, add bias; out is write-once ----
  v8f accs[4] = {acc0, acc1, acc2, acc3};
#pragma unroll
  for (int t = 0; t < 4; ++t) {
    float bb = bc[g * DG + t * 16 + ll];
    v8f a = accs[t];
#pragma unroll
    for (int j = 0; j < 8; ++j) {
      int r = r0 + j + 8 * hl;
      __builtin_nontemporal_store(
          a[j] / lrow[j] + bb,
          out + (size_t)r * FEAT + g * DG + t * 16 + ll);
    }
  }
}

// ---------------------------------------------------------------------------
extern "C" void kernel_launch(void* const* d_in, const int* in_sizes, int n_in,
                              void* d_out, int out_size, void* d_ws, size_t ws_size,
                              hipStream_t stream) {
  const float* pe  = (const float*)d_in[0];   // (R,N,EMB)
  const float* roi = (const float*)d_in[1];   // (R,FEAT)  (== nongt, nongt_dim = N)
  const float* Wq  = (const float*)d_in[2];
  const float* bq  = (const float*)d_in[3];
  const float* Wk  = (const float*)d_in[4];
  const float* bk  = (const float*)d_in[5];
  const float* Wp  = (const float*)d_in[6];
  const float* bp  = (const float*)d_in[7];
  const float* Wc  = (const float*)d_in[8];   // (G,DG,FEAT)
  const float* bc  = (const float*)d_in[9];
  // d_in[10] = nongt_dim == N (compile-time constant here)

  float* ws   = (float*)d_ws;
  float* q    = ws;                  // R*FEAT      = 1M floats
  float* kmat = ws + (1u << 20);     // N*FEAT      = 1M floats
  float* KVt  = ws + (2u << 20);     // G*DG*N      = 1M floats
  float* plog = ws + (3u << 20);     // R*G*N       = 16M floats (64MB)
  float* out  = (float*)d_out;       // (R,FEAT)

  k_proj  <<<256,  256, 0, stream>>>(roi, Wq, bq, Wk, bk, q, kmat);
  k_kv    <<<128,  256, 0, stream>>>(roi, Wc, KVt);
  k_poslog<<<8192, 256, 0, stream>>>(pe, Wp, bp, plog);
  k_attn  <<<128,  256, 0, stream>>>(q, kmat, KVt, plog, bc, out);
}